// ExponentialUnitNorm_30185030156327
// MI455X (gfx1250) — compile-verified
//
#include <hip/hip_runtime.h>
#include <stdint.h>
#include <math.h>

#define ALPHA 0.99f
#define ONEMA 0.01f
#define EPSV  1e-14f
#define L2A   (-0.0144995696f)     // log2(0.99)

constexpr int  BB   = 16;          // batch
constexpr int  TT   = 2000;        // time
constexpr int  FF   = 481;         // freq bins
constexpr int  CH   = 25;          // chunks over T
constexpr int  LCH  = 80;          // chunk length (CH*LCH == TT)
constexpr int  NSUB = 5;           // 16-step sub-chunks per chunk
constexpr int  FT   = (FF + 15) / 16;          // 31 freq tiles
constexpr long TOTAL2 = (long)BB * TT * FF;    // number of float2 samples

typedef float v2f __attribute__((ext_vector_type(2)));
typedef float v8f __attribute__((ext_vector_type(8)));

// ---- single-instruction hardware math (1 ulp is plenty for this op) -------
__device__ __forceinline__ float fast_sqrt(float x) {
#if __has_builtin(__builtin_amdgcn_sqrtf)
    return __builtin_amdgcn_sqrtf(x);
#else
    return sqrtf(x);
#endif
}
__device__ __forceinline__ float fast_rsq(float x) {
#if __has_builtin(__builtin_amdgcn_rsqf)
    return __builtin_amdgcn_rsqf(x);
#else
    return 1.0f / sqrtf(x);
#endif
}
__device__ __forceinline__ float fast_exp2(float x) {
#if __has_builtin(__builtin_amdgcn_exp2f)
    return __builtin_amdgcn_exp2f(x);
#else
    return exp2f(x);
#endif
}

// Low 32 bits of a flat pointer into LDS == wave-relative LDS byte address
// (ISA 10.2).  The ptrtoint also captures the
// shared object so the asm "memory" clobbers are known to write it.
__device__ __forceinline__ uint32_t lds_addr_of(const void* p) {
    return (uint32_t)(uintptr_t)p;
}

// ---------------------------------------------------------------------------
// Stage one 16(time) x 16(freq) tile of float2 x-samples into LDS using CDNA5
// async global->LDS copies. Row t starts are only 8B aligned (F=481), so b64.
// Op q copies rows 2q, 2q+1: lane -> row = 2q + (lane>>4), col = lane&15.
// LDS dest byte = lds_dst + q*256 + lane*8  (tile is row-major 16x16 float2).
// ---------------------------------------------------------------------------
__device__ __forceinline__ void async_stage(const float* x, int b, int t_base,
                                            int f0, int lane, uint32_t lds_dst) {
    const int col = lane & 15, half = lane >> 4;
    const unsigned long long xb = (unsigned long long)(uintptr_t)x;
    unsigned long long g0 = xb +
        8ull * (unsigned long long)(((long)(b * TT + t_base + half)) * FF + f0 + col);
    const unsigned long long gmax = xb + 8ull * (unsigned long long)(TOTAL2 - 1);
    const unsigned long long rstride = 8ull * (unsigned long long)FF;  // one t row
#pragma unroll
    for (int q = 0; q < 8; ++q) {
        unsigned long long g = g0 + (unsigned long long)(2 * q) * rstride;
        g = (g < gmax) ? g : gmax;                 // clamp F-tail overread
        uint32_t l = lds_dst + (uint32_t)(q * 256 + lane * 8);
        asm volatile("global_load_async_to_lds_b64 %0, %1, off"
                     :: "v"(l), "v"(g) : "memory");
    }
}

// ---------------------------------------------------------------------------
// 16-step EMA scan over 16 freq columns as a chained WMMA:
//   S[i][n] = sum_j W[i][j]*m[j][n] + a^(i+1)*s_in[n]
// W[i][j] = (1-a)*a^(i-j) lower-triangular, fed as A in 4 K=4 slices.
// areg[kk][e] = W[lane&15][4kk + 2*(lane>>4) + e]  (A 16x4 f32 VGPR layout)
// pr[v]       = a^(v + 8*(lane>>4) + 1)            (C init, rows v / v+8)
// s_col in/out: state for freq column (lane&15), replicated in both halves.
// ---------------------------------------------------------------------------
__device__ __forceinline__ v8f scan16(const float2* tile,
                                      const float (&areg)[4][2],
                                      const float (&pr)[8],
                                      float& s_col, int lane) {
    const int col = lane & 15, half = lane >> 4;
    v8f c;
#pragma unroll
    for (int v = 0; v < 8; ++v) c[v] = pr[v] * s_col;
#pragma unroll
    for (int kk = 0; kk < 4; ++kk) {
        int k0 = 4 * kk + 2 * half;
        float2 x0 = tile[k0 * 16 + col];
        float2 x1 = tile[(k0 + 1) * 16 + col];
        float  m0 = fast_sqrt(fmaxf(x0.x * x0.x + x0.y * x0.y, EPSV));
        float  m1 = fast_sqrt(fmaxf(x1.x * x1.x + x1.y * x1.y, EPSV));
        v2f a;  a[0] = areg[kk][0]; a[1] = areg[kk][1];
        v2f bm; bm[0] = m0;         bm[1] = m1;
        c = __builtin_amdgcn_wmma_f32_16x16x4_f32(false, a, false, bm,
                                                  (short)0, c, false, false);
    }
    // next s_in = row 15 = c[7] in lanes 16..31
    s_col = __shfl(c[7], 16 + col, 32);
    return c;
}

// Coefficients via one v_exp_f32 each: a^d = exp2(d * log2 a), d in [0,16].
__device__ __forceinline__ void make_coeffs(int lane, float (&areg)[4][2],
                                            float (&pr)[8]) {
    const int col = lane & 15, half = lane >> 4;
#pragma unroll
    for (int kk = 0; kk < 4; ++kk)
#pragma unroll
        for (int e = 0; e < 2; ++e) {
            int j = 4 * kk + 2 * half + e;
            int d = col - j;
            areg[kk][e] = (d >= 0) ? ONEMA * fast_exp2((float)d * L2A) : 0.0f;
        }
#pragma unroll
    for (int v = 0; v < 8; ++v)
        pr[v] = fast_exp2((float)(v + 8 * half + 1) * L2A);
}

// ---------------------------------------------------------------------------
// Kernel 1: per-(b, chunk, ftile) partial P = scan of the chunk with s_in = 0.
// ---------------------------------------------------------------------------
__global__ void __launch_bounds__(32)
k_partials(const float* __restrict__ x, float* __restrict__ P) {
    __shared__ float2 smem[512];                    // 2 x (16x16 float2) tiles
    const uint32_t sbase = lds_addr_of(smem);
    int id = blockIdx.x;
    int ft = id % FT; int rem = id / FT;
    int c  = rem % CH; int b = rem / CH;
    int lane = threadIdx.x;
    int col = lane & 15, half = lane >> 4;
    int f0 = ft * 16, t0 = c * LCH;

    float areg[4][2], pr[8];
    make_coeffs(lane, areg, pr);

    float s_col = 0.0f;
    async_stage(x, b, t0, f0, lane, sbase);
#pragma unroll
    for (int s = 0; s < NSUB; ++s) {
        if (s + 1 < NSUB) {
            async_stage(x, b, t0 + (s + 1) * 16, f0, lane,
                        sbase + (uint32_t)(((s + 1) & 1) * 2048));
            asm volatile("s_wait_asynccnt 0x8" ::: "memory");
        } else {
            asm volatile("s_wait_asynccnt 0x0" ::: "memory");
        }
        (void)scan16(smem + (s & 1) * 256, areg, pr, s_col, lane);
    }
    if (half == 0 && f0 + col < FF)
        P[((long)(b * CH + c)) * FF + f0 + col] = s_col;
}

// ---------------------------------------------------------------------------
// Kernel 2: serial combine over the 25 chunks: s_in(c+1) = a^80*s_in(c) + P_c.
// One thread per (b,f) chain; tiny.
// ---------------------------------------------------------------------------
__global__ void k_combine(const float* __restrict__ P,
                          const float* __restrict__ init_state,
                          float* __restrict__ SIN) {
    int idx = blockIdx.x * blockDim.x + threadIdx.x;
    if (idx >= BB * FF) return;
    int b = idx / FF, f = idx % FF;
    float aL = 1.0f;
    for (int i = 0; i < LCH; ++i) aL *= ALPHA;     // a^80, exact repeated mul
    float s = init_state[f];
    for (int c = 0; c < CH; ++c) {
        long o = ((long)(b * CH + c)) * FF + f;
        SIN[o] = s;
        s = aL * s + P[o];
    }
}

// ---------------------------------------------------------------------------
// Kernel 3: redo the WMMA scan with the true incoming state and emit
// out = x * rsqrt(state).  x re-read (mostly L2-resident: 123MB < 192MB L2).
// ---------------------------------------------------------------------------
__global__ void __launch_bounds__(32)
k_apply(const float* __restrict__ x, const float* __restrict__ SIN,
        float* __restrict__ out) {
    __shared__ float2 smem[512];
    const uint32_t sbase = lds_addr_of(smem);
    int id = blockIdx.x;
    int ft = id % FT; int rem = id / FT;
    int c  = rem % CH; int b = rem / CH;
    int lane = threadIdx.x;
    int col = lane & 15, half = lane >> 4;
    int f0 = ft * 16, t0 = c * LCH;

    float areg[4][2], pr[8];
    make_coeffs(lane, areg, pr);

    int  fc  = f0 + col;
    int  fcl = (fc < FF) ? fc : (FF - 1);
    bool valid = fc < FF;
    float s_col = SIN[((long)(b * CH + c)) * FF + fcl];

    float2* out2 = (float2*)out;

    async_stage(x, b, t0, f0, lane, sbase);
#pragma unroll
    for (int s = 0; s < NSUB; ++s) {
        if (s + 1 < NSUB) {
            async_stage(x, b, t0 + (s + 1) * 16, f0, lane,
                        sbase + (uint32_t)(((s + 1) & 1) * 2048));
            asm volatile("s_wait_asynccnt 0x8" ::: "memory");
        } else {
            asm volatile("s_wait_asynccnt 0x0" ::: "memory");
        }
        const float2* tile = smem + (s & 1) * 256;
        v8f D = scan16(tile, areg, pr, s_col, lane);
#pragma unroll
        for (int v = 0; v < 8; ++v) {
            int row = v + 8 * half;
            int t   = t0 + s * 16 + row;
            float2 xv = tile[row * 16 + col];
            float  inv = fast_rsq(D[v]);
            if (valid) {
                float2 o; o.x = xv.x * inv; o.y = xv.y * inv;
                out2[((long)(b * TT + t)) * FF + fc] = o;
            }
        }
    }
}

// ---------------------------------------------------------------------------
extern "C" void kernel_launch(void* const* d_in, const int* in_sizes, int n_in,
                              void* d_out, int out_size, void* d_ws, size_t ws_size,
                              hipStream_t stream) {
    (void)in_sizes; (void)n_in; (void)out_size; (void)ws_size;
    const float* x    = (const float*)d_in[0];
    const float* init = (const float*)d_in[1];
    float* out = (float*)d_out;

    float* P   = (float*)d_ws;                       // [B*CH*F] partials
    float* SIN = P + (size_t)BB * CH * FF;           // [B*CH*F] incoming states

    dim3 g1(BB * CH * FT);                           // 12,400 one-wave blocks
    k_partials<<<g1, 32, 0, stream>>>(x, P);

    int n2 = BB * FF;
    k_combine<<<(n2 + 255) / 256, 256, 0, stream>>>(P, init, SIN);

    k_apply<<<g1, 32, 0, stream>>>(x, SIN, out);
}